// LabelSmoothingCrossEntropy_IB_70583492542719
// MI455X (gfx1250) — compile-verified
//
#include <hip/hip_runtime.h>
#include <hip/hip_bf16.h>
#include <math.h>

// Problem constants (fixed by the reference)
constexpr int N_ROWS  = 32768;
constexpr int CLS     = 1000;   // CLASS_NUM
constexpr int KDIM    = 512;    // K_IN == K_OUT
constexpr int RPW     = 16;     // rows per wave (matches WMMA M=16)
constexpr int NWAVES  = N_ROWS / RPW;   // 2048

typedef __attribute__((ext_vector_type(2))) float v2f;
typedef __attribute__((ext_vector_type(4))) float v4f;
typedef __attribute__((ext_vector_type(8))) float v8f;

// ---------------------------------------------------------------------------
// Kernel 1: logQt[c*512 + j] = log(Q[j*1000 + c])   (transpose + log, 2 MB)
// ---------------------------------------------------------------------------
__global__ void logq_kernel(const float* __restrict__ Q, float* __restrict__ logQt) {
    int i = blockIdx.x * blockDim.x + threadIdx.x;     // coalesced read of Q
    if (i < KDIM * CLS) {
        int j = i / CLS;
        int c = i - j * CLS;
        logQt[(size_t)c * KDIM + j] = __logf(Q[i]);
    }
}

// ---------------------------------------------------------------------------
// Kernel 2: main fused pass. One wave32 handles 16 consecutive rows.
//  - online softmax over x row (single pass, NT loads)
//  - csi entropy, cso sum
//  - term_2 for the 16 rows via V_WMMA_F32_16X16X4_F32 diagonal trick
//  - per-wave double partials (deterministic; no FP atomics)
// ---------------------------------------------------------------------------
__global__ void __launch_bounds__(256)
main_kernel(const float* __restrict__ x, const int* __restrict__ tgt,
            const float* __restrict__ csi, const float* __restrict__ cso,
            const float* __restrict__ logQt, double* __restrict__ partials)
{
    const int tid  = blockIdx.x * blockDim.x + threadIdx.x;
    const int wave = tid >> 5;
    const int lane = threadIdx.x & 31;
    const int base = wave * RPW;           // grid sized so base < N_ROWS always

    double loss_acc = 0.0;
    double ib_acc   = 0.0;

    for (int r = 0; r < RPW; ++r) {
        const int n = base + r;
        const int t = tgt[n];

        // ---- label-smoothed CE: online softmax + sum over the 1000 logits ----
        const v4f* xrow = reinterpret_cast<const v4f*>(x + (size_t)n * CLS);
        float m = -INFINITY, s = 0.0f, xsum = 0.0f;
        for (int c4 = lane; c4 < CLS / 4; c4 += 32) {
            v4f v = __builtin_nontemporal_load(&xrow[c4]);   // stream-once data
            float m4 = fmaxf(fmaxf(v.x, v.y), fmaxf(v.z, v.w));
            float M  = fmaxf(m, m4);
            s = s * __expf(m - M)
              + __expf(v.x - M) + __expf(v.y - M) + __expf(v.z - M) + __expf(v.w - M);
            m = M;
            xsum += (v.x + v.y) + (v.z + v.w);
        }
        // wave32 reduction of (m, s) online pair and xsum
        for (int off = 16; off; off >>= 1) {
            float m2 = __shfl_xor(m, off, 32);
            float s2 = __shfl_xor(s, off, 32);
            float M  = fmaxf(m, m2);
            s = s * __expf(m - M) + s2 * __expf(m2 - M);
            m = M;
            xsum += __shfl_xor(xsum, off, 32);
        }
        const float lse    = __logf(s) + m;
        const float xt     = x[(size_t)n * CLS + t];          // broadcast gather
        const float nll    = lse - xt;                        // -(logprob at target)
        const float smooth = lse - xsum * (1.0f / CLS);       // -mean(logprobs)
        const float row_loss = 0.9f * nll + 0.1f * smooth;

        // ---- csi entropy: sum csi*log(csi) ----
        const v4f* irow = reinterpret_cast<const v4f*>(csi + (size_t)n * KDIM);
        float ent = 0.0f;
        for (int c4 = lane; c4 < KDIM / 4; c4 += 32) {
            v4f v = __builtin_nontemporal_load(&irow[c4]);
            ent += v.x * __logf(v.x) + v.y * __logf(v.y)
                 + v.z * __logf(v.z) + v.w * __logf(v.w);
        }
        // ---- cso sum ----
        const v4f* orow = reinterpret_cast<const v4f*>(cso + (size_t)n * KDIM);
        float osum = 0.0f;
        for (int c4 = lane; c4 < KDIM / 4; c4 += 32) {
            v4f v = orow[c4];
            osum += (v.x + v.y) + (v.z + v.w);
        }
        for (int off = 16; off; off >>= 1) {
            ent  += __shfl_xor(ent,  off, 32);
            osum += __shfl_xor(osum, off, 32);
        }
        if (lane == 0) {
            loss_acc += (double)row_loss;
            ib_acc   += (double)(osum * ent);   // term_1
        }
    }

    // ---- term_2 for all 16 rows at once via WMMA f32 16x16x4, diag(D) ----
    // A[m][k]  = cso[base+m][k]         (lane&15 = m; lanes>=16 hold K phase +2)
    // B[k][n'] = logQt[t[base+n']][k]   (same lane layout with N in place of M)
    // D[m][n'] accumulates over K=512 in 128 steps; diagonal = wanted dots.
    const int mn      = lane & 15;
    const int kphase  = (lane >> 4) << 1;   // 0 for lanes 0-15, 2 for lanes 16-31
    const int tB      = tgt[base + mn];
    const float* aRow = cso   + (size_t)(base + mn) * KDIM;
    const float* bRow = logQt + (size_t)tB * KDIM;

    v8f dmat = {};
    for (int k0 = 0; k0 < KDIM; k0 += 4) {
        v2f a = *reinterpret_cast<const v2f*>(aRow + k0 + kphase);  // K = k0+kphase, +1
        v2f b = *reinterpret_cast<const v2f*>(bRow + k0 + kphase);
        dmat = __builtin_amdgcn_wmma_f32_16x16x4_f32(
            /*neg_a=*/false, a, /*neg_b=*/false, b,
            /*c_mod=*/(short)0, dmat, /*reuse_a=*/false, /*reuse_b=*/false);
    }

    // Diagonal element (m,m): lane = (m<8 ? m : m+16), vgpr = m & 7.
    float t2 = 0.0f;
#pragma unroll
    for (int mm = 0; mm < 16; ++mm) {
        const int src = (mm < 8) ? mm : (mm + 16);
        float dv = __shfl(dmat[mm & 7], src, 32);
        if (lane == 0) t2 += dv;
    }

    if (lane == 0) {
        ib_acc -= (double)t2;               // term_1 - term_2 summed over 16 rows
        partials[2 * wave + 0] = loss_acc;
        partials[2 * wave + 1] = ib_acc;
    }
}

// ---------------------------------------------------------------------------
// Kernel 3: deterministic final reduction of 2048 double pairs -> 2 floats
// ---------------------------------------------------------------------------
__global__ void finish_kernel(const double* __restrict__ partials, float* __restrict__ out) {
    __shared__ double shl[256];
    __shared__ double shi[256];
    double l = 0.0, ib = 0.0;
    for (int i = threadIdx.x; i < NWAVES; i += 256) {
        l  += partials[2 * i + 0];
        ib += partials[2 * i + 1];
    }
    shl[threadIdx.x] = l;
    shi[threadIdx.x] = ib;
    __syncthreads();
    for (int s = 128; s; s >>= 1) {
        if (threadIdx.x < (unsigned)s) {
            shl[threadIdx.x] += shl[threadIdx.x + s];
            shi[threadIdx.x] += shi[threadIdx.x + s];
        }
        __syncthreads();
    }
    if (threadIdx.x == 0) {
        out[0] = (float)(shl[0] * (1.0 / N_ROWS));
        out[1] = (float)(shi[0] * (1.0 / N_ROWS));
    }
}

// ---------------------------------------------------------------------------
extern "C" void kernel_launch(void* const* d_in, const int* in_sizes, int n_in,
                              void* d_out, int out_size, void* d_ws, size_t ws_size,
                              hipStream_t stream) {
    (void)in_sizes; (void)n_in; (void)out_size; (void)ws_size;

    const float* x   = (const float*)d_in[0];   // [N, 1000]
    const int*   tgt = (const int*)  d_in[1];   // [N]
    const float* csi = (const float*)d_in[2];   // [N, 512]
    const float* cso = (const float*)d_in[3];   // [N, 512]
    const float* Q   = (const float*)d_in[4];   // [512, 1000]
    float* out = (float*)d_out;                 // {loss, IB_Loss}

    // Workspace layout: [0, 2048000) logQt floats ; then 2048 double pairs.
    char*   ws       = (char*)d_ws;
    float*  logQt    = (float*)ws;                          // 512000 * 4 B
    double* partials = (double*)(ws + (size_t)KDIM * CLS * sizeof(float)); // 32 KB

    logq_kernel<<<(KDIM * CLS + 255) / 256, 256, 0, stream>>>(Q, logQt);

    const int threads = NWAVES * 32;                        // 65536
    main_kernel<<<threads / 256, 256, 0, stream>>>(x, tgt, csi, cso, logQt, partials);

    finish_kernel<<<1, 256, 0, stream>>>(partials, out);
}